// OrientationFinder_83193516523885
// MI455X (gfx1250) — compile-verified
//
#include <hip/hip_runtime.h>
#include <math.h>

#define NBINS 36
#define PS 32
#define PATCHES_PER_BLOCK 16
#define THREADS 256
#define WAVES 8

typedef __attribute__((ext_vector_type(2))) float v2f;
typedef __attribute__((ext_vector_type(8))) float v8f;

__device__ __forceinline__ float fast_atan2f(float y, float x) {
    const float PIf   = 3.14159265358979f;
    const float PI_2f = 1.57079632679490f;
    float ax = __builtin_fabsf(x), ay = __builtin_fabsf(y);
    float mx = fmaxf(ax, ay), mn = fminf(ax, ay);
    // t in [0,1]; guard mx==0 (masked pixels) so atan2(0,0)=0 like the reference
    float t = mn * __builtin_amdgcn_rcpf(fmaxf(mx, 1e-30f));
    float s = t * t;
    float p = t * fmaf(s, fmaf(s, fmaf(s, fmaf(s, 0.0208351f, -0.085133f),
                                       0.180141f), -0.3302995f), 0.999866f);
    float r = (ay > ax) ? (PI_2f - p) : p;
    if (x < 0.0f) r = PIf - r;
    return (y < 0.0f) ? -r : r;
}

__global__ __launch_bounds__(THREADS)
void OrientationFinder_kernel(const float* __restrict__ x,
                              const float* __restrict__ gxw,
                              const float* __restrict__ gyw,
                              const float* __restrict__ sw,
                              const float* __restrict__ gk,
                              float* __restrict__ out)
{
    __shared__ float gkS[PS * PS];                       // 4 KB gaussian window
    __shared__ float stageS[WAVES][PS * PS];             // 32 KB patch staging (1 patch/wave)
    __shared__ __align__(16) float histS[16][48];        // 3 KB hist (16 patches x 36 bins, padded)
    __shared__ __align__(16) float smS[16][48];          // 3 KB smoothed hist
    __shared__ __align__(16) float BmT[48][40];          // 7.5 KB smoothing matrix, transposed:
                                                         // BmT[j][k] = M[k][j]

    const int tid  = threadIdx.x;
    const int wave = tid >> 5;
    const int lane = tid & 31;

    const float s0 = sw[0], s1 = sw[1], s2 = sw[2];

    // --- block init: stage gaussian window, zero hist/sm, build smoothing matrix ---
    for (int i = tid; i < PS * PS; i += THREADS) gkS[i] = gk[i];
    for (int i = tid; i < 16 * 48; i += THREADS) {
        (&histS[0][0])[i] = 0.0f;
        (&smS[0][0])[i]   = 0.0f;
    }
    // sm[j] = s0*h[j-1] + s1*h[j] + s2*h[j+1]  (zero padding), j,k in [0,36)
    // BmT[j][k] = s0 if k==j-1, s1 if k==j, s2 if k==j+1, else 0
    for (int i = tid; i < 48 * 40; i += THREADS) {
        const int j = i / 40;
        const int k = i - j * 40;
        float v = 0.0f;
        if (j < NBINS && k < NBINS) {
            if (k == j - 1)      v = s0;
            else if (k == j)     v = s1;
            else if (k == j + 1) v = s2;
        }
        (&BmT[0][0])[i] = v;
    }
    __syncthreads();

    const float gxw0 = gxw[0], gxw1 = gxw[1], gxw2 = gxw[2];
    const float gyw0 = gyw[0], gyw1 = gyw[1], gyw2 = gyw[2];

    const float TWO_PIf   = 6.28318530717959f;
    const float BIN_SCALE = (float)NBINS / TWO_PIf;

    // --- each wave handles 2 patches ---
    for (int i = 0; i < 2; ++i) {
        const int prow = 2 * wave + i;                       // row in hist matrix
        const long long gpatch =
            (long long)blockIdx.x * PATCHES_PER_BLOCK + prow;

        // stage the 4KB patch into this wave's LDS buffer with async copies:
        // 8 x (32 lanes x 16B) = 4096B
        {
            unsigned lds_base = (unsigned)(size_t)(&stageS[wave][0]);
            const char* gbase = (const char*)x + gpatch * (long long)(PS * PS * 4);
            // previous patch's LDS reads must drain before the async engine overwrites
            asm volatile("s_wait_dscnt 0" ::: "memory");
#pragma unroll
            for (int ch = 0; ch < 8; ++ch) {
                unsigned lds_addr = lds_base + ch * 512 + lane * 16;
                unsigned long long ga =
                    (unsigned long long)(gbase + ch * 512 + lane * 16);
                asm volatile("global_load_async_to_lds_b128 %0, %1, off"
                             :: "v"(lds_addr), "v"(ga) : "memory");
            }
            asm volatile("s_wait_asynccnt 0" ::: "memory");
        }

        // lane = column; loop over rows; neighbors read from LDS (conflict-free)
        const int c  = lane;
        const int cl = (c == 0) ? 0 : c - 1;
        const int cr = (c == PS - 1) ? PS - 1 : c + 1;
        const float* st = &stageS[wave][0];
#pragma unroll 4
        for (int r = 0; r < PS; ++r) {
            const int ru = (r == 0) ? 0 : r - 1;
            const int rd = (r == PS - 1) ? PS - 1 : r + 1;
            float cv = st[r * PS + c];
            float lv = st[r * PS + cl];
            float rv = st[r * PS + cr];
            float uv = st[ru * PS + c];
            float dv = st[rd * PS + c];
            float gxv = gxw0 * lv + gxw1 * cv + gxw2 * rv;
            float gyv = gyw0 * uv + gyw1 * cv + gyw2 * dv;
            float mag = __builtin_amdgcn_sqrtf(gxv * gxv + gyv * gyv + 1e-10f)
                        * gkS[r * PS + c];
            float keep = (mag > 0.001f) ? 1.0f : 0.0f;
            gxv *= keep; gyv *= keep; mag *= keep;
            float ori = fast_atan2f(gyv, gxv);
            float th  = (ori < 0.0f) ? ori + TWO_PIf : ori;
            float ob  = th * BIN_SCALE;
            float bf  = floorf(ob);
            float w1  = ob - bf;
            int bin   = (int)bf;
            if (bin >= NBINS) bin -= NBINS;      // handles ob == 36.0 edge
            float wo0 = (1.0f - w1) * mag;
            atomicAdd(&histS[prow][bin], wo0);   // ds_add_f32
        }
    }
    __syncthreads();

    // --- angular smoothing as GEMM: sm(16x36) = hist(16x36) * M(36x36) ---
    // waves 0..2 each compute one 16-wide N tile; 9 K-steps of WMMA f32 16x16x4.
    // A fragment (16x4): lanes 0-15 hold K=k0,k0+1; lanes 16-31 hold K=k0+2,k0+3.
    // B fragment (4x16): VGPR0 rows K=k0 / k0+2, VGPR1 rows K=k0+1 / k0+3.
    // Both fragments are single aligned ds_load_b64 from LDS.
    if (wave < 3) {
        const int half = lane >> 4;       // 0: lanes 0-15, 1: lanes 16-31
        const int m    = lane & 15;       // row (patch) for A; col-in-tile for B/D
        const int jg   = wave * 16 + m;   // global output bin column
        v8f acc = {};
#pragma unroll
        for (int ks = 0; ks < 9; ++ks) {
            const int kb0 = 4 * ks + 2 * half;
            v2f a = *(const v2f*)&histS[m][kb0];
            v2f b = *(const v2f*)&BmT[jg][kb0];
            acc = __builtin_amdgcn_wmma_f32_16x16x4_f32(
                /*neg_a=*/false, a, /*neg_b=*/false, b,
                /*c_mod=*/(short)0, acc, /*reuse_a=*/false, /*reuse_b=*/false);
        }
        // D layout: VGPR g -> row M=g (lanes 0-15) / M=g+8 (lanes 16-31), N=lane&15
#pragma unroll
        for (int g = 0; g < 8; ++g) {
            smS[g + 8 * half][wave * 16 + m] = acc[g];
        }
    }
    __syncthreads();

    // --- argmax over 36 bins per patch (first-max like jnp.argmax), emit angle ---
    if (tid < PATCHES_PER_BLOCK) {
        float best = smS[tid][0];
        int bi = 0;
#pragma unroll
        for (int j = 1; j < NBINS; ++j) {
            float v = smS[tid][j];
            if (v > best) { best = v; bi = j; }
        }
        const float PIf = 3.14159265358979f;
        float angle = -(TWO_PIf * (float)bi / (float)NBINS - PIf);
        out[(long long)blockIdx.x * PATCHES_PER_BLOCK + tid] = angle;
    }
}

extern "C" void kernel_launch(void* const* d_in, const int* in_sizes, int n_in,
                              void* d_out, int out_size, void* d_ws, size_t ws_size,
                              hipStream_t stream) {
    const float* x   = (const float*)d_in[0];
    const float* gxw = (const float*)d_in[1];
    const float* gyw = (const float*)d_in[2];
    const float* sw  = (const float*)d_in[3];
    const float* gk  = (const float*)d_in[4];
    float* out = (float*)d_out;

    const int B = in_sizes[0] / (PS * PS);                 // 65536 patches
    const int grid = (B + PATCHES_PER_BLOCK - 1) / PATCHES_PER_BLOCK;
    OrientationFinder_kernel<<<grid, THREADS, 0, stream>>>(x, gxw, gyw, sw, gk, out);
}